// MultiHeadElman_83829171683679
// MI455X (gfx1250) — compile-verified
//
#include <hip/hip_runtime.h>
#include <hip/hip_bf16.h>

typedef float v2f __attribute__((ext_vector_type(2)));
typedef float v4f __attribute__((ext_vector_type(4)));
typedef float v8f __attribute__((ext_vector_type(8)));

// Problem constants (from reference)
#define TT 4096
#define BB 4
#define DD 2048
#define HH 64
#define HD 32
#define ROWS (TT * BB)   // 16384

// ---------------------------------------------------------------------------
// Phase 1: input projection wx[t,b,h,:] = Wx[h] @ x[t,b,h,:] + bias[h]
// Batched GEMM per head: C[16384,32] = X_h[16384,32] @ Wx[h]^T, via
// V_WMMA_F32_16X16X4_F32. One wave handles a 16-row M-tile, full N=32
// (two v8f accumulators), K=32 as 8 k-steps of 4.
// ---------------------------------------------------------------------------
__global__ __launch_bounds__(256)
void proj_wmma_kernel(const float* __restrict__ x,
                      const float* __restrict__ Wx,
                      const float* __restrict__ bias,
                      float* __restrict__ wx_out)
{
    const int lane = threadIdx.x & 31;
    const int wave = threadIdx.x >> 5;
    const int head = blockIdx.y;                      // 0..63
    const int rowbase = (blockIdx.x * 8 + wave) * 16; // 0..16368 step 16

    const int m  = lane & 15;   // M row within tile (A), N col within tile (B/C)
    const int kh = lane >> 4;   // K half-select (A/B), M half-select (C)

    const float* xrow  = x  + (size_t)(rowbase + m) * DD + head * HD;
    const float* wbase = Wx + (size_t)head * HD * HD;   // Wx[h][i][j], i=out col, j=k

    // Accumulators initialized with bias broadcast down columns.
    float b0 = bias[head * HD + m];
    float b1 = bias[head * HD + 16 + m];
    v8f c0, c1;
#pragma unroll
    for (int v = 0; v < 8; ++v) { c0[v] = b0; c1[v] = b1; }

#pragma unroll
    for (int kk = 0; kk < 8; ++kk) {
        const int col = kk * 4 + kh * 2;  // K pair held by this lane-half
        // A fragment: 16x4 fp32, lane m holds rows, VGPR pair holds K(+lane-half)
        v2f a = *(const v2f*)(xrow + col);
        // B fragments: 4x16 fp32, B[k][n] = Wx[h][n][k]
        v2f bn0 = *(const v2f*)(wbase + (size_t)(m) * HD + col);
        v2f bn1 = *(const v2f*)(wbase + (size_t)(16 + m) * HD + col);
        c0 = __builtin_amdgcn_wmma_f32_16x16x4_f32(false, a, false, bn0,
                                                   (short)0, c0, false, false);
        c1 = __builtin_amdgcn_wmma_f32_16x16x4_f32(false, a, false, bn1,
                                                   (short)0, c1, false, false);
    }

    // C layout: VGPR v, lanes 0-15 -> M=v, lanes 16-31 -> M=v+8; N = lane&15.
    float* obase = wx_out + (size_t)head * HD + m;
#pragma unroll
    for (int v = 0; v < 8; ++v) {
        const size_t row = (size_t)(rowbase + kh * 8 + v);
        obase[row * DD]      = c0[v];
        obase[row * DD + 16] = c1[v];
    }
}

// ---------------------------------------------------------------------------
// Phase 2: sequential Elman scan. One wave per (b,h) chain (256 waves total).
// Lane i owns row i of R[h]; h[j] broadcast per step via v_readlane -> SGPR,
// feeding v_fmac directly. 4 partial accumulators break the FMA chain.
// FUSED=true also applies Wx on the fly (fallback when ws is too small).
// ---------------------------------------------------------------------------
__device__ __forceinline__ float bcast_lane(float v, int j) {
    return __uint_as_float(__builtin_amdgcn_readlane(__float_as_uint(v), j));
}

template <bool FUSED>
__global__ __launch_bounds__(256)
void elman_scan_kernel(const float* __restrict__ src,   // FUSED ? x : wx
                       const float* __restrict__ R,
                       const float* __restrict__ Wx,    // FUSED only
                       const float* __restrict__ bias,  // FUSED only
                       const float* __restrict__ h0,
                       float* __restrict__ out,
                       float* __restrict__ hfinal)
{
    const int lane = threadIdx.x & 31;
    const int wave = (blockIdx.x << 3) + (threadIdx.x >> 5); // 0..255
    const int b = wave >> 6;   // 0..3
    const int h = wave & 63;   // 0..63

    // Row i of R[h] held in registers (32 VGPRs), loaded as b128s.
    float Rr[HD];
    {
        const float* rrow = R + ((size_t)h * HD + lane) * HD;
#pragma unroll
        for (int q = 0; q < 8; ++q) {
            v4f t = *(const v4f*)(rrow + q * 4);
            Rr[q * 4 + 0] = t[0]; Rr[q * 4 + 1] = t[1];
            Rr[q * 4 + 2] = t[2]; Rr[q * 4 + 3] = t[3];
        }
    }

    float Wr[HD];
    float bv = 0.0f;
    if (FUSED) {
        const float* wrow = Wx + ((size_t)h * HD + lane) * HD;
#pragma unroll
        for (int q = 0; q < 8; ++q) {
            v4f t = *(const v4f*)(wrow + q * 4);
            Wr[q * 4 + 0] = t[0]; Wr[q * 4 + 1] = t[1];
            Wr[q * 4 + 2] = t[2]; Wr[q * 4 + 3] = t[3];
        }
        bv = bias[h * HD + lane];
    }

    float hv = h0[((size_t)b * HH + h) * HD + lane];

    const size_t base = (size_t)b * DD + h * HD + lane;   // (t*B+b)*2048 + h*32 + lane
    for (int t = 0; t < TT; ++t) {
        const size_t idx = base + (size_t)t * (BB * DD);
        const float sv = src[idx];     // wx value, or x value when FUSED

        float acc[4];
        acc[0] = FUSED ? bv : sv;
        acc[1] = 0.0f; acc[2] = 0.0f; acc[3] = 0.0f;

#pragma unroll
        for (int j = 0; j < HD; ++j) {
            const float hj = bcast_lane(hv, j);
            acc[j & 3] = fmaf(hj, Rr[j], acc[j & 3]);
        }
        if (FUSED) {
#pragma unroll
            for (int j = 0; j < HD; ++j) {
                const float xj = bcast_lane(sv, j);
                acc[j & 3] = fmaf(xj, Wr[j], acc[j & 3]);
            }
        }

        const float pre = (acc[0] + acc[1]) + (acc[2] + acc[3]);
        hv = pre / (1.0f + fabsf(pre));   // softsign
        out[idx] = hv;
    }

    hfinal[((size_t)b * HH + h) * HD + lane] = hv;
}

// ---------------------------------------------------------------------------
extern "C" void kernel_launch(void* const* d_in, const int* in_sizes, int n_in,
                              void* d_out, int out_size, void* d_ws, size_t ws_size,
                              hipStream_t stream)
{
    const float* x    = (const float*)d_in[0];  // [T,B,D]
    const float* h0   = (const float*)d_in[1];  // [B,H,HD]
    const float* R    = (const float*)d_in[2];  // [H,HD,HD]
    const float* Wx   = (const float*)d_in[3];  // [H,HD,HD]
    const float* bias = (const float*)d_in[4];  // [H,HD]

    float* out    = (float*)d_out;                       // [T,B,D]
    float* hfinal = out + (size_t)TT * BB * DD;          // [B,H,HD] appended

    const size_t wx_bytes = (size_t)ROWS * DD * sizeof(float); // 128 MB

    if (ws_size >= wx_bytes) {
        float* wx_ws = (float*)d_ws;
        // Phase 1: WMMA input projection (memory-bound, ~256 MB traffic).
        proj_wmma_kernel<<<dim3(ROWS / (16 * 8), HH), 256, 0, stream>>>(
            x, Wx, bias, wx_ws);
        // Phase 2: latency-bound sequential scan, 256 chains.
        elman_scan_kernel<false><<<dim3(32), 256, 0, stream>>>(
            wx_ws, R, nullptr, nullptr, h0, out, hfinal);
    } else {
        // Fallback: fused scan computing Wx @ x on the fly.
        elman_scan_kernel<true><<<dim3(32), 256, 0, stream>>>(
            x, R, Wx, bias, h0, out, hfinal);
    }
}